// SignedGraphConvolutionalNetwork_24352464568465
// MI455X (gfx1250) — compile-verified
//
#include <hip/hip_runtime.h>
#include <math.h>

typedef float v2f __attribute__((ext_vector_type(2)));
typedef float v8f __attribute__((ext_vector_type(8)));

#define F_DIM 128
#define H1D 96
#define H2D 64

// ---------------------------------------------------------------------------
// Zero fill (graph-capture safe, no hipMemsetAsync dependence)
// ---------------------------------------------------------------------------
__global__ void zero_kernel(float* __restrict__ p, long long n) {
  long long i = (long long)blockIdx.x * blockDim.x + threadIdx.x;
  long long stride = (long long)gridDim.x * blockDim.x;
  for (; i < n; i += stride) p[i] = 0.0f;
}

// ---------------------------------------------------------------------------
// Base scatter: one wave32 per edge, 128 floats = 4 per lane.
// agg[row] += X[col], cnt[row] += 1 for row != col.
// ---------------------------------------------------------------------------
__global__ void scatter_base(const float* __restrict__ X, const int* __restrict__ ei,
                             float* __restrict__ agg, float* __restrict__ cnt, int E) {
  int gid = blockIdx.x * blockDim.x + threadIdx.x;
  int edge = gid >> 5;
  int lane = gid & 31;
  if (edge >= E) return;
  int row = ei[edge];
  int col = ei[E + edge];
  if (row == col) return;  // uniform per wave
  const float4 v = *(const float4*)(X + (size_t)col * F_DIM + lane * 4);
  float* dst = agg + (size_t)row * F_DIM + lane * 4;
  atomicAdd(dst + 0, v.x);
  atomicAdd(dst + 1, v.y);
  atomicAdd(dst + 2, v.z);
  atomicAdd(dst + 3, v.w);
  if (lane == 0) atomicAdd(cnt + row, 1.0f);
}

// agg[r][j] /= max(cnt[r], 1)
__global__ void div_base(float* __restrict__ agg, const float* __restrict__ cnt,
                         long long total) {
  long long i = (long long)blockIdx.x * blockDim.x + threadIdx.x;
  if (i >= total) return;
  int r = (int)(i / F_DIM);
  agg[i] *= (1.0f / fmaxf(cnt[r], 1.0f));
}

// ---------------------------------------------------------------------------
// Base GEMM: out16x96 = [agg | X](16x256) @ W(256x96) + b, l2norm, tanh.
// 6 waves per block, one 16x16 column tile per wave, f32 WMMA 16x16x4.
// ---------------------------------------------------------------------------
__global__ void __launch_bounds__(192)
base_gemm(const float* __restrict__ agg, const float* __restrict__ X,
          const float* __restrict__ W, const float* __restrict__ bias,
          float* __restrict__ H, int N) {
  __shared__ float smem[16 * H1D];
  __shared__ float sinv[16];
  const int tid = threadIdx.x;
  const int wave = tid >> 5;
  const int lane = tid & 31;
  const int half = lane >> 4;   // 0: K0/K1, 1: K2/K3 (A); same split for B
  const int lr = lane & 15;
  const int colbase = wave * 16;
  const int tile = blockIdx.x;
  int row = tile * 16 + lr;
  if (row >= N) row = N - 1;    // clamp: keep EXEC all-ones for WMMA

  v8f c = {0.f, 0.f, 0.f, 0.f, 0.f, 0.f, 0.f, 0.f};
  for (int k0 = 0; k0 < 2 * F_DIM; k0 += 4) {
    const int kk = k0 + 2 * half;  // quad never crosses the 128 boundary
    const float* src = (kk < F_DIM) ? (agg + (size_t)row * F_DIM + kk)
                                    : (X + (size_t)row * F_DIM + (kk - F_DIM));
    v2f a; a.x = src[0]; a.y = src[1];
    v2f b;
    b.x = W[(size_t)kk * H1D + colbase + lr];
    b.y = W[(size_t)(kk + 1) * H1D + colbase + lr];
    c = __builtin_amdgcn_wmma_f32_16x16x4_f32(false, a, false, b, (short)0, c,
                                              false, false);
  }

  const float bcol = bias[colbase + lr];
#pragma unroll
  for (int v = 0; v < 8; ++v) {
    const int rl = v + 8 * half;                 // C/D layout: M = v + 8*(lane/16)
    smem[rl * H1D + colbase + lr] = c[v] + bcol; // N = lane%16
  }
  __syncthreads();

  if (tid < 16) {
    float s = 0.0f;
    for (int j = 0; j < H1D; ++j) { float x = smem[tid * H1D + j]; s += x * x; }
    sinv[tid] = 1.0f / fmaxf(sqrtf(s), 1e-12f);
  }
  __syncthreads();

  for (int idx = tid; idx < 16 * H1D; idx += 192) {
    const int rl = idx / H1D;
    const int cc = idx - rl * H1D;
    const int gr = tile * 16 + rl;
    if (gr < N) H[(size_t)gr * H1D + cc] = tanhf(smem[idx] * sinv[rl]);
  }
}

// ---------------------------------------------------------------------------
// Deep scatter: one wave per edge, two 96-float payloads (3 per lane each).
// acc1[row] += x1[col]; acc2[row] += x2[col]   (row != col)
// ---------------------------------------------------------------------------
__global__ void scatter_deep(const float* __restrict__ x1, const float* __restrict__ x2,
                             const int* __restrict__ ei,
                             float* __restrict__ acc1, float* __restrict__ acc2, int E) {
  int gid = blockIdx.x * blockDim.x + threadIdx.x;
  int edge = gid >> 5;
  int lane = gid & 31;
  if (edge >= E) return;
  int row = ei[edge];
  int col = ei[E + edge];
  if (row == col) return;
  const float* s1 = x1 + (size_t)col * H1D + lane * 3;
  const float* s2 = x2 + (size_t)col * H1D + lane * 3;
  float* d1 = acc1 + (size_t)row * H1D + lane * 3;
  float* d2 = acc2 + (size_t)row * H1D + lane * 3;
#pragma unroll
  for (int i = 0; i < 3; ++i) atomicAdd(d1 + i, s1[i]);
#pragma unroll
  for (int i = 0; i < 3; ++i) atomicAdd(d2 + i, s2[i]);
}

// acc[r][j] = (acc[r][j] + x[r][j]) / (cnt[r] + 1)   (add self-loop, mean)
__global__ void deep_norm(float* __restrict__ acc, const float* __restrict__ x,
                          const float* __restrict__ cnt, long long total) {
  long long i = (long long)blockIdx.x * blockDim.x + threadIdx.x;
  if (i >= total) return;
  int r = (int)(i / H1D);
  acc[i] = (acc[i] + x[i]) * (1.0f / (cnt[r] + 1.0f));
}

// ---------------------------------------------------------------------------
// Deep GEMM: out16x64 = [B0|B1|B2](16x288) @ W(288x64) + b, l2norm, tanh.
// 4 waves per block; writes into d_out at column offset col_off (stride 128).
// ---------------------------------------------------------------------------
__global__ void __launch_bounds__(128)
deep_gemm(const float* __restrict__ B0, const float* __restrict__ B1,
          const float* __restrict__ B2, const float* __restrict__ W,
          const float* __restrict__ bias, float* __restrict__ OUT,
          int N, int col_off) {
  __shared__ float smem[16 * H2D];
  __shared__ float sinv[16];
  const int tid = threadIdx.x;
  const int wave = tid >> 5;
  const int lane = tid & 31;
  const int half = lane >> 4;
  const int lr = lane & 15;
  const int colbase = wave * 16;
  const int tile = blockIdx.x;
  int row = tile * 16 + lr;
  if (row >= N) row = N - 1;

  v8f c = {0.f, 0.f, 0.f, 0.f, 0.f, 0.f, 0.f, 0.f};
  for (int k0 = 0; k0 < 3 * H1D; k0 += 4) {
    const int kk = k0 + 2 * half;          // region boundaries (96,192) are 4-aligned
    const int region = kk / H1D;
    const int off = kk - region * H1D;
    const float* B = (region == 0) ? B0 : ((region == 1) ? B1 : B2);
    const float* src = B + (size_t)row * H1D + off;
    v2f a; a.x = src[0]; a.y = src[1];
    v2f b;
    b.x = W[(size_t)kk * H2D + colbase + lr];
    b.y = W[(size_t)(kk + 1) * H2D + colbase + lr];
    c = __builtin_amdgcn_wmma_f32_16x16x4_f32(false, a, false, b, (short)0, c,
                                              false, false);
  }

  const float bcol = bias[colbase + lr];
#pragma unroll
  for (int v = 0; v < 8; ++v) {
    const int rl = v + 8 * half;
    smem[rl * H2D + colbase + lr] = c[v] + bcol;
  }
  __syncthreads();

  if (tid < 16) {
    float s = 0.0f;
    for (int j = 0; j < H2D; ++j) { float x = smem[tid * H2D + j]; s += x * x; }
    sinv[tid] = 1.0f / fmaxf(sqrtf(s), 1e-12f);
  }
  __syncthreads();

  for (int idx = tid; idx < 16 * H2D; idx += 128) {
    const int rl = idx / H2D;
    const int cc = idx - rl * H2D;
    const int gr = tile * 16 + rl;
    if (gr < N)
      OUT[(size_t)gr * (2 * H2D) + col_off + cc] = tanhf(smem[idx] * sinv[rl]);
  }
}

// ---------------------------------------------------------------------------
// Host-side orchestration
// ---------------------------------------------------------------------------
extern "C" void kernel_launch(void* const* d_in, const int* in_sizes, int n_in,
                              void* d_out, int out_size, void* d_ws, size_t ws_size,
                              hipStream_t stream) {
  (void)n_in; (void)out_size; (void)ws_size;
  const float* X    = (const float*)d_in[0];
  const int*   ep   = (const int*)d_in[1];
  const int*   en   = (const int*)d_in[2];
  const float* W_bp = (const float*)d_in[3];
  const float* b_bp = (const float*)d_in[4];
  const float* W_bn = (const float*)d_in[5];
  const float* b_bn = (const float*)d_in[6];
  const float* W_dp = (const float*)d_in[7];
  const float* b_dp = (const float*)d_in[8];
  const float* W_dn = (const float*)d_in[9];
  const float* b_dn = (const float*)d_in[10];
  float* out = (float*)d_out;

  const int N = in_sizes[0] / F_DIM;
  const int E = in_sizes[1] / 2;

  // Workspace layout (accA/accB alias the base agg buffers — safe after base GEMMs)
  char* ws = (char*)d_ws;
  const size_t szAgg = (size_t)N * F_DIM * sizeof(float);
  const size_t szH   = (size_t)N * H1D * sizeof(float);
  const size_t szCnt = (((size_t)N * sizeof(float)) + 255) & ~(size_t)255;
  float* aggP = (float*)(ws);
  float* aggN = (float*)(ws + szAgg);
  float* cntP = (float*)(ws + 2 * szAgg);
  float* cntN = (float*)(ws + 2 * szAgg + szCnt);
  float* hP   = (float*)(ws + 2 * szAgg + 2 * szCnt);
  float* hN   = (float*)(ws + 2 * szAgg + 2 * szCnt + szH);
  float* accC = (float*)(ws + 2 * szAgg + 2 * szCnt + 2 * szH);
  float* accD = (float*)(ws + 2 * szAgg + 2 * szCnt + 3 * szH);
  float* accA = aggP;  // reuse
  float* accB = aggN;  // reuse

  const int ZB = 2048, ZT = 256;
  const int edgeBlocks = (E * 32 + 255) / 256;
  const int tiles = (N + 15) / 16;

  // Phase 1: base aggregation (agg buffers + counts are contiguous)
  zero_kernel<<<ZB, ZT, 0, stream>>>((float*)ws,
      (long long)((2 * szAgg + 2 * szCnt) / sizeof(float)));
  scatter_base<<<edgeBlocks, 256, 0, stream>>>(X, ep, aggP, cntP, E);
  scatter_base<<<edgeBlocks, 256, 0, stream>>>(X, en, aggN, cntN, E);
  {
    long long tot = (long long)N * F_DIM;
    int blocks = (int)((tot + 255) / 256);
    div_base<<<blocks, 256, 0, stream>>>(aggP, cntP, tot);
    div_base<<<blocks, 256, 0, stream>>>(aggN, cntN, tot);
  }

  // Phase 2: base GEMMs -> h_pos, h_neg
  base_gemm<<<tiles, 192, 0, stream>>>(aggP, X, W_bp, b_bp, hP, N);
  base_gemm<<<tiles, 192, 0, stream>>>(aggN, X, W_bn, b_bn, hN, N);

  // Phase 3: deep aggregation (accA/accB reuse agg region; zero all four)
  zero_kernel<<<ZB, ZT, 0, stream>>>((float*)ws, (long long)((2 * szAgg) / sizeof(float)));
  zero_kernel<<<ZB, ZT, 0, stream>>>(accC, (long long)(2 * (szH / sizeof(float))));
  // ep: accA += h_pos[col], accC += h_neg[col]
  scatter_deep<<<edgeBlocks, 256, 0, stream>>>(hP, hN, ep, accA, accC, E);
  // en: accB += h_neg[col], accD += h_pos[col]
  scatter_deep<<<edgeBlocks, 256, 0, stream>>>(hN, hP, en, accB, accD, E);
  {
    long long tot = (long long)N * H1D;
    int blocks = (int)((tot + 255) / 256);
    deep_norm<<<blocks, 256, 0, stream>>>(accA, hP, cntP, tot);
    deep_norm<<<blocks, 256, 0, stream>>>(accB, hN, cntN, tot);
    deep_norm<<<blocks, 256, 0, stream>>>(accC, hN, cntP, tot);
    deep_norm<<<blocks, 256, 0, stream>>>(accD, hP, cntN, tot);
  }

  // Phase 4: deep GEMMs -> out[:, 0:64] and out[:, 64:128]
  deep_gemm<<<tiles, 128, 0, stream>>>(accA, accB, hP, W_dp, b_dp, out, N, 0);
  deep_gemm<<<tiles, 128, 0, stream>>>(accC, accD, hN, W_dn, b_dn, out, N, H2D);
}